// TranAttLayer_83837761618627
// MI455X (gfx1250) — compile-verified
//
#include <hip/hip_runtime.h>
#include <hip/hip_bf16.h>
#include <math.h>

typedef __attribute__((ext_vector_type(2))) float v2f;
typedef __attribute__((ext_vector_type(4))) float v4f;
typedef __attribute__((ext_vector_type(8))) float v8f;

#define BB 2
#define NN 4096
#define DD 256

// ---------------------------------------------------------------------------
// Kernel 0: fold a1/a2 through Wa:  va1[d] = sum_e a1[e]*Wa[e*D+d]
// One block of 256 threads; Wa reads are fully coalesced row by row.
// ---------------------------------------------------------------------------
__global__ __launch_bounds__(DD) void make_va_kernel(
    const float* __restrict__ Wa, const float* __restrict__ a1,
    const float* __restrict__ a2, float* __restrict__ va1,
    float* __restrict__ va2) {
  int d = threadIdx.x;
  float acc1 = 0.f, acc2 = 0.f;
  for (int e = 0; e < DD; ++e) {
    float w = Wa[e * DD + d];
    acc1 = fmaf(a1[e], w, acc1);
    acc2 = fmaf(a2[e], w, acc2);
  }
  va1[d] = acc1;
  va2[d] = acc2;
}

// ---------------------------------------------------------------------------
// Kernel 1: fused GEMV pass via V_WMMA_F32_16X16X4_F32.
//   [B*N, 256] x [256, 16]  where only cols {0,1,2} = {Wd, va1, va2} are live.
// 256 threads = 8 waves; each wave owns a 16-row tile, K loop 256/4 = 64 WMMAs.
// Outputs: demands (sigmoid applied, -> d_out), s, t (-> workspace).
// ---------------------------------------------------------------------------
__global__ __launch_bounds__(256) void gemv_wmma_kernel(
    const float* __restrict__ embed, const float* __restrict__ Wd,
    const float* __restrict__ bd, const float* __restrict__ va1,
    const float* __restrict__ va2, float* __restrict__ demands_out,
    float* __restrict__ s_out, float* __restrict__ t_out) {
  __shared__ float Bt[DD * 16];  // Bt[k*16 + c] : B-matrix table, 16 KB

  int tid = threadIdx.x;
  for (int idx = tid; idx < DD * 16; idx += 256) {
    int k = idx >> 4;
    int cc = idx & 15;
    float v = 0.f;
    if (cc == 0) v = Wd[k];
    else if (cc == 1) v = va1[k];
    else if (cc == 2) v = va2[k];
    Bt[idx] = v;
  }
  __syncthreads();

  int wave  = tid >> 5;
  int lane  = tid & 31;
  int col   = lane & 15;   // A-row index == B-col index == lane&15 (ISA layout)
  int khalf = lane >> 4;   // lanes 16-31 carry K = {2,3} of each K-quad

  long row0 = ((long)blockIdx.x * 8 + wave) * 16;
  const float* arow = embed + (row0 + col) * DD;

  v8f acc = {};
  for (int kb = 0; kb < DD; kb += 4) {
    int klo = kb + khalf * 2;
    v2f a = *(const v2f*)(arow + klo);           // A: row=col, K={klo,klo+1}
    v2f b;
    b.x = Bt[klo * 16 + col];                    // B: col, K=klo
    b.y = Bt[(klo + 1) * 16 + col];              // B: col, K=klo+1
    acc = __builtin_amdgcn_wmma_f32_16x16x4_f32(
        /*neg_a=*/false, a, /*neg_b=*/false, b,
        /*c_mod=*/(short)0, acc, /*reuse_a=*/false, /*reuse_b=*/false);
  }

  // C/D layout: acc[r] at lane L is element (M = r + 8*(L>>4), N = L&15).
  if (col < 3) {
    float bd0 = bd[0];
    long mbase = row0 + (long)khalf * 8;
    #pragma unroll
    for (int r = 0; r < 8; ++r) {
      float v = acc[r];
      long m = mbase + r;
      if (col == 0) {
        demands_out[m] = 1.f / (1.f + __expf(-(v + bd0)));
      } else if (col == 1) {
        s_out[m] = v;
      } else {
        t_out[m] = v;
      }
    }
  }
}

// ---------------------------------------------------------------------------
// Kernel 1b: tmax[b] = max_j t[b, j]   (one block per batch)
// ---------------------------------------------------------------------------
__global__ __launch_bounds__(256) void tmax_kernel(const float* __restrict__ t,
                                                   float* __restrict__ tmax) {
  __shared__ float red[256];
  const float* tb = t + (long)blockIdx.x * NN;
  float m = -INFINITY;
  for (int j = threadIdx.x; j < NN; j += 256) m = fmaxf(m, tb[j]);
  red[threadIdx.x] = m;
  __syncthreads();
  for (int s = 128; s > 0; s >>= 1) {
    if (threadIdx.x < s) red[threadIdx.x] = fmaxf(red[threadIdx.x], red[threadIdx.x + s]);
    __syncthreads();
  }
  if (threadIdx.x == 0) tmax[blockIdx.x] = red[0];
}

// ---------------------------------------------------------------------------
// Kernel 2: one workgroup per output row i of Gs.
//   Gs[b,i,j] = demands[b,i] * exp(lrelu(s_i + t_j) - m_i) / Z_i
//   m_i = lrelu(s_i + tmax_b) (exact row max; lrelu is monotone).
// exp values kept in registers (4 x float4 per thread), single pass + reduce,
// then non-temporal float4 stores (write-once 134 MB stream).
// ---------------------------------------------------------------------------
__global__ __launch_bounds__(256) void softmax_row_kernel(
    const float* __restrict__ s, const float* __restrict__ t,
    const float* __restrict__ demands, const float* __restrict__ tmax,
    float* __restrict__ Gs) {
  long row = blockIdx.x;          // 0 .. B*N-1
  int b = (int)(row >> 12);       // N = 4096
  int tid = threadIdx.x;

  float si = s[row];
  float di = demands[row];
  float m = si + tmax[b];
  m = (m > 0.f) ? m : 0.01f * m;

  const v4f* t4 = (const v4f*)(t + (long)b * NN);
  v4f ev[4];
  float lsum = 0.f;
  #pragma unroll
  for (int it = 0; it < 4; ++it) {
    v4f tv = t4[it * 256 + tid];
    v4f e;
    #pragma unroll
    for (int c = 0; c < 4; ++c) {
      float x = si + tv[c];
      float a = (x > 0.f) ? x : 0.01f * x;     // leaky_relu, slope 0.01
      e[c] = __expf(a - m);
    }
    ev[it] = e;
    lsum += e[0] + e[1] + e[2] + e[3];
  }

  __shared__ float red[256];
  red[tid] = lsum;
  __syncthreads();
  for (int st = 128; st > 0; st >>= 1) {
    if (tid < st) red[tid] += red[tid + st];
    __syncthreads();
  }
  float scale = di / red[0];

  v4f* g4 = (v4f*)(Gs + row * (long)NN);
  #pragma unroll
  for (int it = 0; it < 4; ++it) {
    v4f o = ev[it] * scale;
    __builtin_nontemporal_store(o, g4 + it * 256 + tid);
  }
}

// ---------------------------------------------------------------------------
extern "C" void kernel_launch(void* const* d_in, const int* in_sizes, int n_in,
                              void* d_out, int out_size, void* d_ws, size_t ws_size,
                              hipStream_t stream) {
  const float* embed = (const float*)d_in[0];  // [B, N, D]
  const float* Wd    = (const float*)d_in[1];  // [D]
  const float* bd    = (const float*)d_in[2];  // [1]
  const float* Wa    = (const float*)d_in[3];  // [D, D]
  const float* a1    = (const float*)d_in[4];  // [D]
  const float* a2    = (const float*)d_in[5];  // [D]

  float* out      = (float*)d_out;
  float* demands  = out;            // [B, N]      (first 8192 floats)
  float* Gs       = out + BB * NN;  // [B, N, N]

  float* ws   = (float*)d_ws;
  float* va1  = ws;                 // [256]
  float* va2  = ws + DD;            // [256]
  float* sbuf = ws + 2 * DD;        // [B*N]
  float* tbuf = sbuf + BB * NN;     // [B*N]
  float* tmax = tbuf + BB * NN;     // [B]

  make_va_kernel<<<1, DD, 0, stream>>>(Wa, a1, a2, va1, va2);

  // B*N = 8192 rows, 16 rows/wave, 8 waves/block -> 64 blocks
  gemv_wmma_kernel<<<(BB * NN) / (16 * 8), 256, 0, stream>>>(
      embed, Wd, bd, va1, va2, demands, sbuf, tbuf);

  tmax_kernel<<<BB, 256, 0, stream>>>(tbuf, tmax);

  softmax_row_kernel<<<BB * NN, 256, 0, stream>>>(sbuf, tbuf, demands, tmax, Gs);
}